// CircleLoss_23038204575781
// MI455X (gfx1250) — compile-verified
//
#include <hip/hip_runtime.h>
#include <hip/hip_bf16.h>
#include <math.h>

// Circle loss, fused: S = E*E^T via V_WMMA_F32_16X16X4_F32 with the B panel
// double-buffered in LDS via CDNA5 async global->LDS copies (ASYNCcnt).
// Online masked logsumexp per anchor row; S (64MB) never materialized.

typedef __attribute__((ext_vector_type(2))) float v2f;
typedef __attribute__((ext_vector_type(8))) float v8f;

#define MARGIN   0.4f
#define GAMMA    80.0f
#define NEG_INF  (-1e30f)
#define WAVES_PER_BLOCK 8
#define BLOCK_THREADS   (WAVES_PER_BLOCK * 32)
#define COL_PAD         132              // 128 + 4 pad floats: conflict-free ds_load_b64
#define PANEL_FLOATS    (16 * COL_PAD)   // one 16-col x 128-K panel

// Online logsumexp update: state (m, s) represents lse = m + log(s).
__device__ __forceinline__ void lse_update(float& m, float& s, float x) {
    float mn = fmaxf(m, x);
    s = s * __expf(m - mn) + __expf(x - mn);
    m = mn;
}

__device__ __forceinline__ void lse_combine(float& m, float& s, float m2, float s2) {
    float mn = fmaxf(m, m2);
    s = s * __expf(m - mn) + s2 * __expf(m2 - mn);
    m = mn;
}

__global__ __launch_bounds__(BLOCK_THREADS)
void circle_rows_kernel(const float* __restrict__ E,
                        const int*   __restrict__ labels,
                        const int*   __restrict__ batch_ptr,
                        float* __restrict__ per_anchor,   // [B]
                        float* __restrict__ valid_flag,   // [B]
                        int B, int D) {
    __shared__ float sh[2 * PANEL_FLOATS];

    const int tid  = threadIdx.x;
    const int lane = tid & 31;                  // wave32
    const int wave = tid >> 5;
    const int num_tiles = B >> 4;

    int row_tile = blockIdx.x * WAVES_PER_BLOCK + wave;
    const bool active = (row_tile < num_tiles);
    if (!active) row_tile = num_tiles - 1;      // clamp; still join fills/barriers
    const int row_base = row_tile << 4;
    const int hi  = lane >> 4;
    const int lid = lane & 15;
    const int batch = batch_ptr[0];

    // Raw LDS offset of the shared buffer (low 32 bits of generic pointer).
    const unsigned lds0 = (unsigned)(uintptr_t)(&sh[0]);

    // ---- Cooperative async fill: 256 threads x 8 floats = one 16x128 panel.
    const int fcol = tid >> 4;                  // 0..15  (column in tile)
    const int fk   = (tid & 15) << 3;           // 0,8,...,120 (K offset)
    const unsigned fill_lds0 = lds0 + (unsigned)((fcol * COL_PAD + fk) * 4);
    const unsigned fill_lds1 = fill_lds0 + (unsigned)(PANEL_FLOATS * 4);

    auto issue_fill = [&](int tile) {
        const float* g = E + (size_t)((tile << 4) + fcol) * D + fk;
        const unsigned ldsa = (tile & 1) ? fill_lds1 : fill_lds0;
        const unsigned long long ga = (unsigned long long)(uintptr_t)g;
        // INST_OFFSET applies to both LDS and global address: our layout is
        // contiguous on both sides within a column, so offset:16 is correct.
        asm volatile(
            "global_load_async_to_lds_b128 %0, %1, off\n\t"
            "global_load_async_to_lds_b128 %0, %1, off offset:16"
            :: "v"(ldsa), "v"(ga) : "memory");
    };

    // ---- Preload A panel: 16 rows x 128 K, resident for whole kernel.
    // A layout (16x4 f32): M = lane&15; lanes 0-15 K={0,1}, lanes 16-31 K={2,3}.
    const float* arow = E + (size_t)(row_base + lid) * D + 2 * hi;
    v2f areg[32];
#pragma unroll
    for (int t = 0; t < 32; ++t)
        areg[t] = *(const v2f*)(arow + 4 * t);

    // Row labels for this lane's 8 C rows (M = r + 8*hi).
    int lab_row[8];
#pragma unroll
    for (int r = 0; r < 8; ++r)
        lab_row[r] = labels[row_base + r + 8 * hi];

    // Online LSE state + pair counts per C slot.
    float mp[8], sp[8], mn_[8], sn[8], cp[8], cn[8];
#pragma unroll
    for (int r = 0; r < 8; ++r) {
        mp[r] = NEG_INF; sp[r] = 0.f;
        mn_[r] = NEG_INF; sn[r] = 0.f;
        cp[r] = 0.f; cn[r] = 0.f;
    }

    const float OP = 1.0f + MARGIN, DP = 1.0f - MARGIN;
    const float DN = MARGIN;

    // ---- Software pipeline: prime buffer 0.
    issue_fill(0);
    asm volatile("s_wait_asynccnt 0x0" ::: "memory");
    __syncthreads();

    for (int tile = 0; tile < num_tiles; ++tile) {
        const int cur = tile & 1;
        if (tile + 1 < num_tiles) issue_fill(tile + 1);  // overlaps with compute

        const int mycol = (tile << 4) + lid;
        const int lab_c = labels[mycol];

        // Burst-read B pairs from LDS (conflict-free, see COL_PAD).
        const float* bpanel = &sh[cur * PANEL_FLOATS + lid * COL_PAD + 2 * hi];
        v2f breg[32];
#pragma unroll
        for (int t = 0; t < 32; ++t)
            breg[t] = *(const v2f*)(bpanel + 4 * t);

        // Two independent accumulation chains to halve WMMA RAW depth.
        v8f c0 = {}, c1 = {};
#pragma unroll
        for (int t = 0; t < 16; ++t) {
            c0 = __builtin_amdgcn_wmma_f32_16x16x4_f32(
                     false, areg[2 * t],     false, breg[2 * t],
                     (short)0, c0, false, false);
            c1 = __builtin_amdgcn_wmma_f32_16x16x4_f32(
                     false, areg[2 * t + 1], false, breg[2 * t + 1],
                     (short)0, c1, false, false);
        }

        // C/D layout: VGPR r -> row M = r + 8*hi, col N = lid.
#pragma unroll
        for (int r = 0; r < 8; ++r) {
            const float sv   = c0[r] + c1[r];
            const int   rowg = row_base + r + 8 * hi;
            const bool  same   = (lab_row[r] == lab_c);
            const bool  is_pos = same && (rowg != mycol);
            const bool  is_neg = !same;

            const float ap = fmaxf(OP - sv, 0.0f);       // alpha_p (stop-grad)
            const float lp = -GAMMA * ap * (sv - DP);    // logit_p
            const float an = fmaxf(sv + MARGIN, 0.0f);   // alpha_n
            const float ln = GAMMA * an * (sv - DN);     // logit_n

            lse_update(mp[r],  sp[r], is_pos ? lp : NEG_INF);
            lse_update(mn_[r], sn[r], is_neg ? ln : NEG_INF);
            cp[r] += is_pos ? 1.0f : 0.0f;
            cn[r] += is_neg ? 1.0f : 0.0f;
        }

        // Fill(t+1) complete (per-wave) + cross-wave visibility / reuse safety.
        asm volatile("s_wait_asynccnt 0x0" ::: "memory");
        __syncthreads();
    }

    // ---- Reduce across the 16 lanes of each half-wave (over N).
#pragma unroll
    for (int off = 1; off < 16; off <<= 1) {
#pragma unroll
        for (int r = 0; r < 8; ++r) {
            float m2 = __shfl_xor(mp[r], off, 32);
            float s2 = __shfl_xor(sp[r], off, 32);
            lse_combine(mp[r], sp[r], m2, s2);
            m2 = __shfl_xor(mn_[r], off, 32);
            s2 = __shfl_xor(sn[r], off, 32);
            lse_combine(mn_[r], sn[r], m2, s2);
            cp[r] += __shfl_xor(cp[r], off, 32);
            cn[r] += __shfl_xor(cn[r], off, 32);
        }
    }

    // ---- Per-anchor finalize: lane 0 -> rows 0..7, lane 16 -> rows 8..15.
    if (active && lid == 0) {
#pragma unroll
        for (int r = 0; r < 8; ++r) {
            const int   a  = row_base + r + 8 * hi;
            const float pc = cp[r], nc = cn[r];
            const bool keep  = ((a % 4 == 0) && (a < batch)) || (a > batch);
            const bool valid = keep && (pc > 0.f) && (nc > 0.f);

            const float lse_p = mp[r]  + __logf(sp[r]) + __logf(fmaxf(nc, 1.f));
            const float lse_n = mn_[r] + __logf(sn[r]) + __logf(fmaxf(pc, 1.f));
            const float z   = lse_n + lse_p;
            const float spz = (z > 20.f) ? z : log1pf(__expf(z)); // softplus
            per_anchor[a] = valid ? spz : 0.f;
            valid_flag[a] = valid ? 1.f : 0.f;
        }
    }
}

// Deterministic final reduction (no float atomics -> bitwise reproducible).
__global__ __launch_bounds__(256)
void circle_reduce_kernel(const float* __restrict__ per_anchor,
                          const float* __restrict__ valid_flag,
                          float* __restrict__ out, int B) {
    __shared__ float ssum[256];
    __shared__ float scnt[256];
    const int t = threadIdx.x;
    float a = 0.f, c = 0.f;
    for (int i = t; i < B; i += 256) {
        a += per_anchor[i];
        c += valid_flag[i];
    }
    ssum[t] = a; scnt[t] = c;
    __syncthreads();
    for (int s = 128; s > 0; s >>= 1) {
        if (t < s) { ssum[t] += ssum[t + s]; scnt[t] += scnt[t + s]; }
        __syncthreads();
    }
    if (t == 0) {
        const float nv = fmaxf(scnt[0], 1.f);
        out[0] = ssum[0] / nv;
    }
}

extern "C" void kernel_launch(void* const* d_in, const int* in_sizes, int n_in,
                              void* d_out, int out_size, void* d_ws, size_t ws_size,
                              hipStream_t stream) {
    const float* E         = (const float*)d_in[0];
    const int*   labels    = (const int*)d_in[1];
    const int*   batch_ptr = (const int*)d_in[2];

    const int B = in_sizes[1];          // 4096 anchors
    const int D = in_sizes[0] / B;      // 128

    float* per_anchor = (float*)d_ws;
    float* valid_flag = per_anchor + B;

    const int num_tiles = B / 16;
    const int blocks = (num_tiles + WAVES_PER_BLOCK - 1) / WAVES_PER_BLOCK;

    circle_rows_kernel<<<blocks, BLOCK_THREADS, 0, stream>>>(
        E, labels, batch_ptr, per_anchor, valid_flag, B, D);
    circle_reduce_kernel<<<1, 256, 0, stream>>>(
        per_anchor, valid_flag, (float*)d_out, B);
}